// TopologicalHypergraphConvLayer_816043786314
// MI455X (gfx1250) — compile-verified
//
#include <hip/hip_runtime.h>
#include <hip/hip_bf16.h>

typedef __attribute__((ext_vector_type(2))) float v2f;
typedef __attribute__((ext_vector_type(8))) float v8f;

#define D 128
#define LDA 132   // padded LDS row (132*4B = 528B, 16B aligned, avoids 16-way bank conflicts)

// ---------------------------------------------------------------------------
// Y[MxD] = X[MxD] @ W[DxD] + bias,  via V_WMMA_F32_16X16X4_F32.
// grid.x = M/16 (row tile), grid.y = 2 (64-col group), block = 128 (4 waves).
// Each wave owns one 16x16 output tile.
// ---------------------------------------------------------------------------
__global__ void gemm128_bias(const float* __restrict__ X, const float* __restrict__ W,
                             const float* __restrict__ bias, float* __restrict__ Y) {
    __shared__ float ldsA[16 * LDA];
    const int tid  = threadIdx.x;
    const int wave = tid >> 5;
    const int lane = tid & 31;
    const int row0 = blockIdx.x << 4;
    const int col0 = (blockIdx.y << 6) + (wave << 4);

    // Stage the 16x128 A tile into LDS (shared by all 4 waves).
    {
        const float4* src = (const float4*)(X + (size_t)row0 * D);
        for (int i = tid; i < 16 * (D / 4); i += 128) {
            const int r = i >> 5, c4 = i & 31;
            ((float4*)(ldsA + r * LDA))[c4] = src[(r << 5) + c4];
        }
    }
    __syncthreads();

    const int m    = lane & 15;   // A row / B col within tile
    const int half = lane >> 4;   // selects K pair {0,1} vs {2,3}
    v8f c = {};
#pragma unroll
    for (int k = 0; k < D; k += 4) {
        v2f a, b;
        const int kk = k + (half << 1);
        a.x = ldsA[m * LDA + kk];
        a.y = ldsA[m * LDA + kk + 1];
        b.x = W[(size_t)kk * D + col0 + m];
        b.y = W[(size_t)(kk + 1) * D + col0 + m];
        c = __builtin_amdgcn_wmma_f32_16x16x4_f32(false, a, false, b, (short)0, c, false, false);
    }

    const float bc = bias[col0 + m];
#pragma unroll
    for (int v = 0; v < 8; ++v) {
        const int mm = v + (half << 3);
        Y[(size_t)(row0 + mm) * D + col0 + m] = c[v] + bc;
    }
}

// ---------------------------------------------------------------------------
// Hyperedge attention: one block = 16 hyperedges, 8 waves cover all 128
// flattened (head, hidden) columns with WMMA; then per-head reduce ->
// sigmoid -> fusion, one scalar per hyperedge.
// aw1 is [H=4][D=128][HD=32]; flattened column c = h*32 + kk.
// ---------------------------------------------------------------------------
__global__ void he_attn(const float* __restrict__ he, const float* __restrict__ aw1,
                        const float* __restrict__ ab1, const float* __restrict__ aw2,
                        const float* __restrict__ ab2, const float* __restrict__ fusion,
                        float* __restrict__ attn) {
    __shared__ float ldsA[16 * LDA];
    __shared__ float ldsH[16 * LDA];
    __shared__ float ldsS[16];
    const int tid  = threadIdx.x;       // 256 threads = 8 waves
    const int wave = tid >> 5;
    const int lane = tid & 31;
    const int row0 = blockIdx.x << 4;

    if (tid < 16) ldsS[tid] = 0.0f;
    {
        const float4* src = (const float4*)(he + (size_t)row0 * D);
        for (int i = tid; i < 16 * (D / 4); i += 256) {
            const int r = i >> 5, c4 = i & 31;
            ((float4*)(ldsA + r * LDA))[c4] = src[(r << 5) + c4];
        }
    }
    __syncthreads();

    const int m    = lane & 15;
    const int half = lane >> 4;
    const int col  = (wave << 4) + m;   // 0..127 flattened (head, hidden)
    const int h    = col >> 5;
    const int hk   = col & 31;
    v8f c = {};
#pragma unroll
    for (int k = 0; k < D; k += 4) {
        v2f a, b;
        const int kk = k + (half << 1);
        a.x = ldsA[m * LDA + kk];
        a.y = ldsA[m * LDA + kk + 1];
        b.x = aw1[h * (D * 32) + kk * 32 + hk];
        b.y = aw1[h * (D * 32) + (kk + 1) * 32 + hk];
        c = __builtin_amdgcn_wmma_f32_16x16x4_f32(false, a, false, b, (short)0, c, false, false);
    }

    const float b1 = ab1[col];
    const float w2 = aw2[col];
#pragma unroll
    for (int v = 0; v < 8; ++v) {
        const int mm = v + (half << 3);
        float hv = c[v] + b1;
        hv = hv > 0.0f ? hv : 0.0f;          // relu
        ldsH[mm * LDA + col] = hv * w2;      // pre-scale by attn_w2
    }
    __syncthreads();

    if (tid < 64) {                          // (row, head) pairs
        const int r = tid >> 2, hh = tid & 3;
        float s = 0.0f;
#pragma unroll
        for (int k2 = 0; k2 < 32; ++k2) s += ldsH[r * LDA + (hh << 5) + k2];
        const float av = 1.0f / (1.0f + __expf(-(s + ab2[hh])));
        atomicAdd(&ldsS[r], av * fusion[hh]);
    }
    __syncthreads();
    if (tid < 16) attn[row0 + tid] = ldsS[tid];
}

// ---------------------------------------------------------------------------
// Scatter: protein -> hyperedge.  32 threads per edge, 4 dims per thread.
// ---------------------------------------------------------------------------
__global__ void scatter_p2he(const float* __restrict__ feat, const int* __restrict__ src,
                             const int* __restrict__ dst, const float* __restrict__ w,
                             float* __restrict__ he, int E) {
    const int t = blockIdx.x * 256 + threadIdx.x;
    const int e = t >> 5;
    if (e >= E) return;
    const int cbase = (t & 31) << 2;
    const float wt = w[e];
    const float4 f = *(const float4*)(feat + (size_t)src[e] * D + cbase);
    float* o = he + (size_t)dst[e] * D + cbase;
    atomicAdd(o + 0, f.x * wt);
    atomicAdd(o + 1, f.y * wt);
    atomicAdd(o + 2, f.z * wt);
    atomicAdd(o + 3, f.w * wt);
}

// ---------------------------------------------------------------------------
// Scatter: hyperedge (scaled by edge attention) -> protein.
// ---------------------------------------------------------------------------
__global__ void scatter_he2p(const float* __restrict__ he, const float* __restrict__ attn,
                             const int* __restrict__ src, const int* __restrict__ dst,
                             const float* __restrict__ w, float* __restrict__ agg, int E) {
    const int t = blockIdx.x * 256 + threadIdx.x;
    const int e = t >> 5;
    if (e >= E) return;
    const int cbase = (t & 31) << 2;
    const int d = dst[e];
    const float sc = w[e] * attn[d];
    const float4 f = *(const float4*)(he + (size_t)d * D + cbase);
    float* o = agg + (size_t)src[e] * D + cbase;
    atomicAdd(o + 0, f.x * sc);
    atomicAdd(o + 1, f.y * sc);
    atomicAdd(o + 2, f.z * sc);
    atomicAdd(o + 3, f.w * sc);
}

__global__ void zero_f4(float4* __restrict__ p, int n4) {
    const int i = blockIdx.x * 256 + threadIdx.x;
    if (i < n4) p[i] = make_float4(0.f, 0.f, 0.f, 0.f);
}

// ---------------------------------------------------------------------------
// Final fusion: pooled means -> pointwise 3x3 conv -> sigmoid gates ->
// weighted sum + residual -> relu.  One block = one protein row.
// ---------------------------------------------------------------------------
__global__ void fuse(const float* __restrict__ feat, const float* __restrict__ f0,
                     const float* __restrict__ f1, const float* __restrict__ f2,
                     const float* __restrict__ cw, float* __restrict__ out) {
    __shared__ float s0[D], s1[D], s2[D];
    const int r = blockIdx.x, d = threadIdx.x;
    const size_t idx = (size_t)r * D + d;
    const float a = f0[idx], b = f1[idx], c = f2[idx];
    s0[d] = a; s1[d] = b; s2[d] = c;
    __syncthreads();
    for (int st = D / 2; st > 0; st >>= 1) {
        if (d < st) { s0[d] += s0[d + st]; s1[d] += s1[d + st]; s2[d] += s2[d + st]; }
        __syncthreads();
    }
    const float p0 = s0[0] * (1.0f / D);
    const float p1 = s1[0] * (1.0f / D);
    const float p2 = s2[0] * (1.0f / D);
    const float g0 = 1.0f / (1.0f + __expf(-(p0 * cw[0] + p1 * cw[1] + p2 * cw[2])));
    const float g1 = 1.0f / (1.0f + __expf(-(p0 * cw[3] + p1 * cw[4] + p2 * cw[5])));
    const float g2 = 1.0f / (1.0f + __expf(-(p0 * cw[6] + p1 * cw[7] + p2 * cw[8])));
    const float fu = a * g0 + b * g1 + c * g2 + feat[idx];
    out[idx] = fu > 0.0f ? fu : 0.0f;
}

extern "C" void kernel_launch(void* const* d_in, const int* in_sizes, int n_in,
                              void* d_out, int out_size, void* d_ws, size_t ws_size,
                              hipStream_t stream) {
    const float* feat = (const float*)d_in[0];
    const int*   src1 = (const int*)d_in[1];
    const int*   dst1 = (const int*)d_in[2];
    const float* w1   = (const float*)d_in[3];
    const int*   src2 = (const int*)d_in[4];
    const int*   dst2 = (const int*)d_in[5];
    const float* w2   = (const float*)d_in[6];
    const float* Ws   = (const float*)d_in[7];
    const float* bs   = (const float*)d_in[8];
    const float* Wf   = (const float*)d_in[9];
    const float* bf   = (const float*)d_in[10];
    const float* Wsec = (const float*)d_in[11];
    const float* bsec = (const float*)d_in[12];
    const float* aw1  = (const float*)d_in[13];
    const float* ab1  = (const float*)d_in[14];
    const float* aw2  = (const float*)d_in[15];
    const float* ab2  = (const float*)d_in[16];
    const float* fusw = (const float*)d_in[17];
    const float* cw   = (const float*)d_in[18];
    float* out = (float*)d_out;

    const int NP  = in_sizes[0] / D;   // 200000
    const int E1  = in_sizes[1];       // 1000000
    const int E2  = in_sizes[4];
    const int NHE = 50000;

    // Workspace layout (f32): feat0 | feat1 | feat2 | agg | he | attn
    float* feat0 = (float*)d_ws;
    float* feat1 = feat0 + (size_t)NP * D;
    float* feat2 = feat1 + (size_t)NP * D;
    float* agg   = feat2 + (size_t)NP * D;
    float* he    = agg + (size_t)NP * D;
    float* attn  = he + (size_t)NHE * D;

    const dim3 gemmGrid(NP / 16, 2);
    const int heQ  = NHE * (D / 4);
    const int aggQ = NP * (D / 4);

    // feat_0 = feat @ Ws + bs
    gemm128_bias<<<gemmGrid, 128, 0, stream>>>(feat, Ws, bs, feat0);

    // --- propagate over edge type 1 -> feat_1 ---
    zero_f4<<<(heQ + 255) / 256, 256, 0, stream>>>((float4*)he, heQ);
    scatter_p2he<<<(E1 * 32 + 255) / 256, 256, 0, stream>>>(feat, src1, dst1, w1, he, E1);
    he_attn<<<NHE / 16, 256, 0, stream>>>(he, aw1, ab1, aw2, ab2, fusw, attn);
    zero_f4<<<(aggQ + 255) / 256, 256, 0, stream>>>((float4*)agg, aggQ);
    scatter_he2p<<<(E1 * 32 + 255) / 256, 256, 0, stream>>>(he, attn, src1, dst1, w1, agg, E1);
    gemm128_bias<<<gemmGrid, 128, 0, stream>>>(agg, Wf, bf, feat1);

    // --- propagate over edge type 2 -> feat_2 ---
    zero_f4<<<(heQ + 255) / 256, 256, 0, stream>>>((float4*)he, heQ);
    scatter_p2he<<<(E2 * 32 + 255) / 256, 256, 0, stream>>>(feat, src2, dst2, w2, he, E2);
    he_attn<<<NHE / 16, 256, 0, stream>>>(he, aw1, ab1, aw2, ab2, fusw, attn);
    zero_f4<<<(aggQ + 255) / 256, 256, 0, stream>>>((float4*)agg, aggQ);
    scatter_he2p<<<(E2 * 32 + 255) / 256, 256, 0, stream>>>(he, attn, src2, dst2, w2, agg, E2);
    gemm128_bias<<<gemmGrid, 128, 0, stream>>>(agg, Wsec, bsec, feat2);

    // --- source-attention fusion + residual + relu ---
    fuse<<<NP, D, 0, stream>>>(feat, feat0, feat1, feat2, cw, out);
}